// MultiHeadAttention_49744311222489
// MI455X (gfx1250) — compile-verified
//
#include <hip/hip_runtime.h>

// Problem constants (match reference setup_inputs)
#define B_     4
#define P_     512
#define H_     512
#define HEADS_ 8
#define D_     64            // H_/HEADS_
#define SCALE_ 0.125f        // 1/sqrt(64)
#define NEGINF_ (-1e9f)

typedef float v2f __attribute__((ext_vector_type(2)));
typedef float v8f __attribute__((ext_vector_type(8)));

// D = A(16x4) * B(4x16) + C, full fp32 (CDNA5 V_WMMA_F32_16X16X4_F32)
__device__ __forceinline__ v8f wmma4(v2f a, v2f b, v8f c) {
  return __builtin_amdgcn_wmma_f32_16x16x4_f32(false, a, false, b, (short)0, c,
                                               false, false);
}

// -------------------------------------------------------------------------
// K1: role-stacked projection.  out[row,o] = x[row,:] . w[role][o,:] + b[role][o]
// rows = b*P+p (2048), cols = H (512).  blockIdx.z = role; write masked on
// roles[p]==role so every element is written exactly once.
// add_emb: for the q projection also add role_emb[role][o % 64].
__global__ void __launch_bounds__(32)
proj_kernel(const float* __restrict__ x, const float* __restrict__ w,
            const float* __restrict__ bias, const int* __restrict__ roles,
            const float* __restrict__ role_emb, float* __restrict__ outp,
            int add_emb) {
  const int lane = threadIdx.x, half = lane >> 4, lq = lane & 15;
  const int role = blockIdx.z;
  const int colBase = blockIdx.x * 16;
  const int rowBase = blockIdx.y * 16;
  const int koff = 2 * half;

  const float* arow = x + (size_t)(rowBase + lq) * H_;
  const float* brow = w + (size_t)role * H_ * H_ + (size_t)(colBase + lq) * H_;

  v8f c = {};
#pragma unroll 4
  for (int kk = 0; kk < H_; kk += 4) {
    v2f a = *(const v2f*)(arow + kk + koff);
    v2f b = *(const v2f*)(brow + kk + koff);
    c = wmma4(a, b, c);
  }

  const int col = colBase + lq;
  const float bv = bias[role * H_ + col];
  const float emb = add_emb ? role_emb[role * D_ + (col & (D_ - 1))] : 0.0f;
#pragma unroll
  for (int i = 0; i < 8; ++i) {
    int row = rowBase + i + 8 * half;
    int p = row & (P_ - 1);
    if (roles[p] == role)
      outp[(size_t)row * H_ + col] = c[i] + bv + emb;
  }
}

// -------------------------------------------------------------------------
// K2a: T1 = Q.K^T per (b,h).  scores[b,h,q,k] = sum_d q[b,q,h,d]*k[b,k,h,d]
__global__ void __launch_bounds__(32)
scores_qk_kernel(const float* __restrict__ qp, const float* __restrict__ kp,
                 float* __restrict__ scores) {
  const int lane = threadIdx.x, half = lane >> 4, lq = lane & 15;
  const int kBase = blockIdx.x * 16, qBase = blockIdx.y * 16;
  const int bh = blockIdx.z, b = bh >> 3, h = bh & 7;
  const int koff = 2 * half;

  const float* arow = qp + (size_t)(b * P_ + qBase + lq) * H_ + h * D_;
  const float* brow = kp + (size_t)(b * P_ + kBase + lq) * H_ + h * D_;

  v8f c = {};
#pragma unroll
  for (int dd = 0; dd < D_; dd += 4) {
    v2f a = *(const v2f*)(arow + dd + koff);
    v2f bf = *(const v2f*)(brow + dd + koff);
    c = wmma4(a, bf, c);
  }
  float* srow = scores + ((size_t)bh * P_ + qBase) * P_ + kBase + lq;
#pragma unroll
  for (int i = 0; i < 8; ++i)
    srow[(size_t)(i + 8 * half) * P_] = c[i];
}

// -------------------------------------------------------------------------
// K2b: T2 (+fused T4) per (b,q):  M=heads(pad16), N=k tile, K=d.
//   T2[h,k] = sum_d q[b,q,h,d] * ek[b,q,k,d]
//   T4[k]   = sum_d eq[b,k,q,d] * ek[b,q,k,d]   (broadcast over h)
__global__ void __launch_bounds__(32)
scores_edge_k_kernel(const float* __restrict__ qp, const float* __restrict__ ek,
                     const float* __restrict__ eq, float* __restrict__ scores) {
  const int lane = threadIdx.x, half = lane >> 4, lq = lane & 15;
  const int kBase = blockIdx.x * 16;
  const int q = blockIdx.y, b = blockIdx.z;
  const int koff = 2 * half;
  const v2f zero2 = {0.0f, 0.0f};

  const float* arow = qp + (size_t)(b * P_ + q) * H_ + lq * D_;          // h = lq
  const float* brow = ek + ((size_t)(b * P_ + q) * P_ + kBase + lq) * D_; // k = kBase+lq
  const bool avalid = lq < 8;

  v8f c = {};
#pragma unroll
  for (int dd = 0; dd < D_; dd += 4) {
    v2f a = avalid ? *(const v2f*)(arow + dd + koff) : zero2;
    v2f bf = *(const v2f*)(brow + dd + koff);
    c = wmma4(a, bf, c);
  }

  // T4 for this lane's column (computed redundantly in both halves; only
  // half==0 writes)
  const float* eqrow = eq + ((size_t)(b * P_ + kBase + lq) * P_ + q) * D_;
  float t4 = 0.0f;
#pragma unroll
  for (int dd = 0; dd < D_; dd += 4) {
    float4 ea = *(const float4*)(eqrow + dd);
    float4 eb = *(const float4*)(brow + dd);
    t4 = fmaf(ea.x, eb.x, t4);
    t4 = fmaf(ea.y, eb.y, t4);
    t4 = fmaf(ea.z, eb.z, t4);
    t4 = fmaf(ea.w, eb.w, t4);
  }

  if (half == 0) {  // rows 0..7 of D hold the 8 valid heads
#pragma unroll
    for (int i = 0; i < 8; ++i)
      scores[(((size_t)(b * HEADS_ + i) * P_) + q) * P_ + kBase + lq] +=
          c[i] + t4;
  }
}

// -------------------------------------------------------------------------
// K2c: T3 per (b,k):  M=q tile, N=heads(pad16), K=d.
//   T3[q,h] = sum_d eq[b,k,q,d] * k[b,k,h,d]
__global__ void __launch_bounds__(32)
scores_edge_q_kernel(const float* __restrict__ kp, const float* __restrict__ eq,
                     float* __restrict__ scores) {
  const int lane = threadIdx.x, half = lane >> 4, lq = lane & 15;
  const int qBase = blockIdx.x * 16;
  const int k = blockIdx.y, b = blockIdx.z;
  const int koff = 2 * half;
  const v2f zero2 = {0.0f, 0.0f};

  const float* arow = eq + ((size_t)(b * P_ + k) * P_ + qBase + lq) * D_; // q row
  const float* brow = kp + (size_t)(b * P_ + k) * H_ + lq * D_;          // h = lq
  const bool bvalid = lq < 8;

  v8f c = {};
#pragma unroll
  for (int dd = 0; dd < D_; dd += 4) {
    v2f a = *(const v2f*)(arow + dd + koff);
    v2f bf = bvalid ? *(const v2f*)(brow + dd + koff) : zero2;
    c = wmma4(a, bf, c);
  }
  if (lq < 8) {
    const int h = lq;
#pragma unroll
    for (int i = 0; i < 8; ++i) {
      int qrow = qBase + i + 8 * half;
      scores[(((size_t)(b * HEADS_ + h) * P_) + qrow) * P_ + k] += c[i];
    }
  }
}

// -------------------------------------------------------------------------
// K3: masked, scaled softmax over k (in place).  mask = any(graph[b,:,q,k]>0)
__global__ void __launch_bounds__(256)
softmax_kernel(const int* __restrict__ graph, float* __restrict__ scores) {
  __shared__ float red[256];
  const int bhq = blockIdx.x;            // (b*8+h)*512 + q
  const int q = bhq & (P_ - 1);
  const int b = (bhq >> 9) >> 3;
  float* row = scores + (size_t)bhq * P_;
  const int* g = graph + ((size_t)b * 4 * P_ + q) * P_;
  const size_t gch = (size_t)P_ * P_;
  const int t = threadIdx.x;

  float vals[2];
#pragma unroll
  for (int j = 0; j < 2; ++j) {
    int k = t + j * 256;
    bool m = (g[k] > 0) || (g[gch + k] > 0) || (g[2 * gch + k] > 0) ||
             (g[3 * gch + k] > 0);
    vals[j] = m ? row[k] * SCALE_ : NEGINF_;
  }
  red[t] = fmaxf(vals[0], vals[1]);
  __syncthreads();
  for (int s = 128; s > 0; s >>= 1) {
    if (t < s) red[t] = fmaxf(red[t], red[t + s]);
    __syncthreads();
  }
  float mx = red[0];
  __syncthreads();
  float e0 = __expf(vals[0] - mx), e1 = __expf(vals[1] - mx);
  red[t] = e0 + e1;
  __syncthreads();
  for (int s = 128; s > 0; s >>= 1) {
    if (t < s) red[t] += red[t + s];
    __syncthreads();
  }
  float inv = 1.0f / red[0];
  row[t] = e0 * inv;
  row[t + 256] = e1 * inv;
}

// -------------------------------------------------------------------------
// K4a: x[b,q,h,d] = sum_k attn[b,h,q,k] * v[b,k,h,d]   (GEMM per b,h)
__global__ void __launch_bounds__(32)
av_kernel(const float* __restrict__ attn, const float* __restrict__ vp,
          float* __restrict__ xb) {
  const int lane = threadIdx.x, half = lane >> 4, lq = lane & 15;
  const int nBase = blockIdx.x * 16, qBase = blockIdx.y * 16;
  const int bh = blockIdx.z, b = bh >> 3, h = bh & 7;
  const int koff = 2 * half;

  const float* arow = attn + ((size_t)bh * P_ + qBase + lq) * P_;
  const float* bcol = vp + (size_t)b * P_ * H_ + h * D_ + nBase + lq;

  v8f c = {};
  for (int kk = 0; kk < P_; kk += 4) {
    v2f a = *(const v2f*)(arow + kk + koff);
    v2f bf;
    bf.x = bcol[(size_t)(kk + koff) * H_];
    bf.y = bcol[(size_t)(kk + koff + 1) * H_];
    c = wmma4(a, bf, c);
  }
#pragma unroll
  for (int i = 0; i < 8; ++i) {
    int qrow = qBase + i + 8 * half;
    xb[(size_t)(b * P_ + qrow) * H_ + h * D_ + nBase + lq] = c[i];
  }
}

// -------------------------------------------------------------------------
// K4b: x[b,q,h,d] += sum_k attn[b,h,q,k] * ev[b,q,k,d]   (per b,q; M=heads pad16)
__global__ void __launch_bounds__(32)
aev_kernel(const float* __restrict__ attn, const float* __restrict__ ev,
           float* __restrict__ xb) {
  const int lane = threadIdx.x, half = lane >> 4, lq = lane & 15;
  const int nBase = blockIdx.x * 16;
  const int q = blockIdx.y, b = blockIdx.z;
  const int koff = 2 * half;
  const v2f zero2 = {0.0f, 0.0f};

  const float* arow = attn + (((size_t)(b * HEADS_ + lq) * P_) + q) * P_; // h=lq
  const bool avalid = lq < 8;
  const float* bcol = ev + (size_t)(b * P_ + q) * P_ * D_ + nBase + lq;

  v8f c = {};
  for (int kk = 0; kk < P_; kk += 4) {
    v2f a = avalid ? *(const v2f*)(arow + kk + koff) : zero2;
    v2f bf;
    bf.x = bcol[(size_t)(kk + koff) * D_];
    bf.y = bcol[(size_t)(kk + koff + 1) * D_];
    c = wmma4(a, bf, c);
  }
  if (half == 0) {
#pragma unroll
    for (int i = 0; i < 8; ++i)
      xb[(size_t)(b * P_ + q) * H_ + i * D_ + nBase + lq] += c[i];
  }
}

// -------------------------------------------------------------------------
// K5: out = xb @ out_w^T + out_b
__global__ void __launch_bounds__(32)
outproj_kernel(const float* __restrict__ xb, const float* __restrict__ ow,
               const float* __restrict__ ob, float* __restrict__ out) {
  const int lane = threadIdx.x, half = lane >> 4, lq = lane & 15;
  const int colBase = blockIdx.x * 16, rowBase = blockIdx.y * 16;
  const int koff = 2 * half;

  const float* arow = xb + (size_t)(rowBase + lq) * H_;
  const float* brow = ow + (size_t)(colBase + lq) * H_;

  v8f c = {};
#pragma unroll 4
  for (int kk = 0; kk < H_; kk += 4) {
    v2f a = *(const v2f*)(arow + kk + koff);
    v2f bf = *(const v2f*)(brow + kk + koff);
    c = wmma4(a, bf, c);
  }
  const int col = colBase + lq;
  const float bv = ob[col];
#pragma unroll
  for (int i = 0; i < 8; ++i)
    out[(size_t)(rowBase + i + 8 * half) * H_ + col] = c[i] + bv;
}

// -------------------------------------------------------------------------
extern "C" void kernel_launch(void* const* d_in, const int* in_sizes, int n_in,
                              void* d_out, int out_size, void* d_ws,
                              size_t ws_size, hipStream_t stream) {
  (void)in_sizes; (void)n_in; (void)out_size; (void)ws_size;
  const float* query = (const float*)d_in[0];
  const float* key   = (const float*)d_in[1];
  const float* value = (const float*)d_in[2];
  const int*   graph = (const int*)d_in[3];
  const float* ek    = (const float*)d_in[4];
  const float* ev    = (const float*)d_in[5];
  const float* eq    = (const float*)d_in[6];
  const int*   roles = (const int*)d_in[7];
  const float* q_w = (const float*)d_in[8];
  const float* q_b = (const float*)d_in[9];
  const float* k_w = (const float*)d_in[10];
  const float* k_b = (const float*)d_in[11];
  const float* v_w = (const float*)d_in[12];
  const float* v_b = (const float*)d_in[13];
  const float* role_emb = (const float*)d_in[14];
  const float* out_w = (const float*)d_in[15];
  const float* out_b = (const float*)d_in[16];
  float* out = (float*)d_out;

  // Workspace layout (floats): qp/kp/vp (3 x 1M), scores (8M), xb (1M) = 48MB
  float* ws = (float*)d_ws;
  const size_t BPH = (size_t)B_ * P_ * H_;
  float* qp = ws;
  float* kp = qp + BPH;
  float* vp = kp + BPH;
  float* scores = vp + BPH;
  float* xb = scores + (size_t)B_ * HEADS_ * P_ * P_;

  dim3 blk(32);

  // 1) role-selected projections (z = role)
  dim3 gProj(H_ / 16, (B_ * P_) / 16, 3);
  proj_kernel<<<gProj, blk, 0, stream>>>(query, q_w, q_b, roles, role_emb, qp, 1);
  proj_kernel<<<gProj, blk, 0, stream>>>(key,   k_w, k_b, roles, role_emb, kp, 0);
  proj_kernel<<<gProj, blk, 0, stream>>>(value, v_w, v_b, roles, role_emb, vp, 0);

  // 2) score terms (stream-ordered: store, then two accumulating passes)
  dim3 gQK(P_ / 16, P_ / 16, B_ * HEADS_);
  scores_qk_kernel<<<gQK, blk, 0, stream>>>(qp, kp, scores);
  dim3 gEdge(P_ / 16, P_, B_);
  scores_edge_k_kernel<<<gEdge, blk, 0, stream>>>(qp, ek, eq, scores);
  scores_edge_q_kernel<<<gEdge, blk, 0, stream>>>(kp, eq, scores);

  // 3) masked scaled softmax (in place on scores -> attn)
  softmax_kernel<<<dim3(B_ * HEADS_ * P_), dim3(256), 0, stream>>>(graph, scores);

  // 4) attention-weighted values: store then accumulate edge term
  dim3 gAV(D_ / 16, P_ / 16, B_ * HEADS_);
  av_kernel<<<gAV, blk, 0, stream>>>(scores, vp, xb);
  dim3 gAEV(D_ / 16, P_, B_);
  aev_kernel<<<gAEV, blk, 0, stream>>>(scores, ev, xb);

  // 5) output projection
  outproj_kernel<<<dim3(H_ / 16, (B_ * P_) / 16), blk, 0, stream>>>(xb, out_w,
                                                                   out_b, out);
}